// YOLO_loss_30855045054997
// MI455X (gfx1250) — compile-verified
//
#include <hip/hip_runtime.h>

typedef __attribute__((ext_vector_type(2))) float v2f;
typedef __attribute__((ext_vector_type(8))) float v8f;

// ---------------------------------------------------------------------------
// Exact 32-lane wave sum using V_WMMA_F32_16X16X4_F32.
// A (16x4 f32, 2 VGPRs): VGPR0 holds {lanes 0-15 -> (M=lane,K=0),
// lanes 16-31 -> (M=lane-16,K=2)}. We place the per-lane partial in VGPR0 and
// zero in VGPR1 (K=1/K=3). With B = ones(4x16):
//   D[m][n] = s[m] + s[m+16]   (all 16 columns identical)
// D layout: VGPR j, lanes 0-15 -> M=j, lanes 16-31 -> M=j+8. Summing the 8 D
// registers gives lanes<16: sum(s[0..7],s[16..23]); lanes>=16: the rest.
// One shfl_xor(16) completes the exact wave total on every lane.
// ---------------------------------------------------------------------------
__device__ __forceinline__ float wave_reduce_wmma(float s) {
  v2f a; a.x = s;    a.y = 0.0f;
  v2f b; b.x = 1.0f; b.y = 1.0f;
  v8f c = {};
  v8f d = __builtin_amdgcn_wmma_f32_16x16x4_f32(
      /*neg_a=*/false, a, /*neg_b=*/false, b,
      /*c_mod=*/(short)0, c, /*reuse_a=*/false, /*reuse_b=*/false);
  float r = ((d[0] + d[1]) + (d[2] + d[3])) + ((d[4] + d[5]) + (d[6] + d[7]));
  r += __shfl_xor(r, 16, 32);
  return r;
}

// Deterministic 256-thread block sum; result valid on threadIdx.x == 0.
__device__ __forceinline__ float block_reduce_256(float s) {
  __shared__ float wsum[8];
  float r = wave_reduce_wmma(s);
  const int lane = threadIdx.x & 31;
  const int wave = threadIdx.x >> 5;
  if (lane == 0) wsum[wave] = r;
  __syncthreads();
  float t = 0.0f;
  if (threadIdx.x == 0) {
    t = ((wsum[0] + wsum[1]) + (wsum[2] + wsum[3])) +
        ((wsum[4] + wsum[5]) + (wsum[6] + wsum[7]));
  }
  return t;
}

constexpr int kCells     = 65536 * 7 * 7;     // 3,211,264
constexpr int kBlockSize = 256;
constexpr int kPerThread = 4;
constexpr int kBlocks    = kCells / (kBlockSize * kPerThread);  // 3136 (exact)

// ---------------------------------------------------------------------------
// Pass 1: each thread handles 4 cells (block-contiguous for coalescing).
// Per cell per tensor: 2x b64 + 1x b32 (base = cell*120 B is 8B-aligned).
// Streaming data -> nontemporal loads (no reuse; 770 MB > 192 MB L2).
// ---------------------------------------------------------------------------
__global__ __launch_bounds__(kBlockSize)
void yolo_loss_partial(const float* __restrict__ pre,
                       const float* __restrict__ gt,
                       float* __restrict__ partial) {
  const int base = blockIdx.x * (kBlockSize * kPerThread) + threadIdx.x;
  float s = 0.0f;
#pragma unroll
  for (int k = 0; k < kPerThread; ++k) {
    const long long cell = (long long)(base + k * kBlockSize);
    const float* p = pre + cell * 30;
    const float* g = gt  + cell * 30;

    v2f p01 = __builtin_nontemporal_load(reinterpret_cast<const v2f*>(p));
    v2f p23 = __builtin_nontemporal_load(reinterpret_cast<const v2f*>(p + 2));
    float p4 = __builtin_nontemporal_load(p + 4);
    v2f g01 = __builtin_nontemporal_load(reinterpret_cast<const v2f*>(g));
    v2f g23 = __builtin_nontemporal_load(reinterpret_cast<const v2f*>(g + 2));
    float g4 = __builtin_nontemporal_load(g + 4);

    v2f d01 = p01 - g01;
    v2f d23 = p23 - g23;
    float coord = (d01.x * d01.x + d01.y * d01.y) +
                  (d23.x * d23.x + d23.y * d23.y);
    float dc   = p4 - g4;
    float conf = dc * dc;
    // gt[...,4] is exactly 0.0f or 1.0f, so exact compare matches reference.
    float cellLoss = (g4 == 1.0f) ? fmaf(5.0f, coord, 10.0f * conf)
                                  : 0.5f * conf;
    s += cellLoss;
  }

  float t = block_reduce_256(s);
  if (threadIdx.x == 0) partial[blockIdx.x] = t;
}

// ---------------------------------------------------------------------------
// Pass 2: one block deterministically folds the 3136 partials into d_out[0].
// ---------------------------------------------------------------------------
__global__ __launch_bounds__(kBlockSize)
void yolo_loss_final(const float* __restrict__ partial, int n,
                     float* __restrict__ out) {
  float s = 0.0f;
  for (int i = threadIdx.x; i < n; i += kBlockSize) s += partial[i];
  float t = block_reduce_256(s);
  if (threadIdx.x == 0) out[0] = t;
}

extern "C" void kernel_launch(void* const* d_in, const int* in_sizes, int n_in,
                              void* d_out, int out_size, void* d_ws, size_t ws_size,
                              hipStream_t stream) {
  const float* pre = (const float*)d_in[0];   // label_pre        [B,7,7,30]
  const float* gt  = (const float*)d_in[1];   // label_ground_truth
  float* partial   = (float*)d_ws;            // kBlocks floats (12.5 KB)
  float* out       = (float*)d_out;

  yolo_loss_partial<<<kBlocks, kBlockSize, 0, stream>>>(pre, gt, partial);
  yolo_loss_final<<<1, kBlockSize, 0, stream>>>(partial, kBlocks, out);
}